// PolyConv_39608188404451
// MI455X (gfx1250) — compile-verified
//
#include <hip/hip_runtime.h>

#define FDIM 64
#define BLOCK 256
#define NODES_PER_BLOCK 32
#define LEN_CAP 2048           // staged CSR entries per block (mean ~512, ~48 sigma margin)
#define SRCM 0x3FFFFF          // src packed in meta[21:0] (N < 2^22), slot in meta[27:22]

typedef float v2f __attribute__((ext_vector_type(2)));
typedef float v8f __attribute__((ext_vector_type(8)));

// ---------------------------------------------------------------------------
__global__ __launch_bounds__(BLOCK) void poly_zero_int(int* __restrict__ p, int n) {
    int i = blockIdx.x * BLOCK + threadIdx.x;
    if (i < n) p[i] = 0;
}

__global__ __launch_bounds__(BLOCK) void poly_degree(const int* __restrict__ eidx,
                                                     int* __restrict__ deg, int E) {
    int e = blockIdx.x * BLOCK + threadIdx.x;
    if (e < E) atomicAdd(&deg[eidx[E + e]], 1);
}

// 3-phase single-block scan: segment sums -> 1024-wide scan -> re-walk emit
__global__ __launch_bounds__(1024) void poly_scan(const int* __restrict__ deg,
                                                  int* __restrict__ rowstart, int n) {
    __shared__ int sums[1024];
    const int t = threadIdx.x;
    const int C = (n + 1023) >> 10;
    const int b0 = t * C;
    const int b1 = (b0 + C < n) ? b0 + C : n;
    int s = 0;
    for (int i = b0; i < b1; ++i) s += deg[i];
    sums[t] = s;
    __syncthreads();
    for (int off = 1; off < 1024; off <<= 1) {
        int add = (t >= off) ? sums[t - off] : 0;
        __syncthreads();
        sums[t] += add;
        __syncthreads();
    }
    int run = (t == 0) ? 0 : sums[t - 1];
    if (t == 0) rowstart[0] = 0;
    for (int i = b0; i < b1; ++i) {
        run += deg[i];
        rowstart[i + 1] = run;
    }
}

__global__ __launch_bounds__(BLOCK) void poly_prep(const int* __restrict__ deg,
                                                   const int* __restrict__ rowstart,
                                                   float* __restrict__ dis,
                                                   int* __restrict__ cursor, int n) {
    int i = blockIdx.x * BLOCK + threadIdx.x;
    if (i < n) {
        dis[i]    = rsqrtf(fmaxf((float)deg[i], 1.0f));
        cursor[i] = rowstart[i];
    }
}

__global__ __launch_bounds__(BLOCK) void poly_build(const int* __restrict__ eidx,
                                                    int* __restrict__ cursor,
                                                    int* __restrict__ csr_src, int E) {
    int e = blockIdx.x * BLOCK + threadIdx.x;
    if (e < E) {
        int s = eidx[e], d = eidx[E + e];
        int pos = atomicAdd(&cursor[d], 1);
        csr_src[pos] = s;
    }
}

__global__ __launch_bounds__(BLOCK) void poly_init_out(const float4* __restrict__ x,
                                                       float4* __restrict__ out,
                                                       float theta0, int n4) {
    int i = blockIdx.x * BLOCK + threadIdx.x;
    if (i < n4) {
        float4 v = x[i];
        out[i] = make_float4(theta0 * v.x, theta0 * v.y, theta0 * v.z, theta0 * v.w);
    }
}

// ---------------------------------------------------------------------------
// One pull-based Laplacian pass with WMMA aggregation and fused finalize.
// Block = 32 nodes; 8 waves = 2 node-groups x 4 feature-quarters.
// Inner loop (branch-free, EXEC all-ones): per 4 edges,
//   1x ds_load_b128 (2 packed records), 2x global b32 gather,
//   B built in regs, 1x v_wmma_f32_16x16x4_f32 accumulate.
__global__ __launch_bounds__(BLOCK) void poly_pass(const int* __restrict__ rowstart,
                                                   const int* __restrict__ csr_src,
                                                   const float* __restrict__ dis,
                                                   const float* __restrict__ fprev,
                                                   float* __restrict__ fcur,
                                                   float* __restrict__ out,
                                                   float theta, int N_) {
    __shared__ int ds_rs[NODES_PER_BLOCK + 1];
    __shared__ int ds_raw[LEN_CAP];                          // staged csr_src slice
    __shared__ __align__(16) int ds_rec[2 * (LEN_CAP + 8)];  // {src|slot<<22, f32 scale}

    const int tid = threadIdx.x;
    const int n0  = blockIdx.x * NODES_PER_BLOCK;

    if (tid <= NODES_PER_BLOCK) {
        int idx = n0 + tid; if (idx > N_) idx = N_;
        ds_rs[tid] = rowstart[idx];
    }
    __syncthreads();

    const int e0 = ds_rs[0];
    int len = ds_rs[NODES_PER_BLOCK] - e0;
    if (len > LEN_CAP) len = LEN_CAP;

    // gfx1250 async copy: stage this block's contiguous CSR slice into LDS
    for (int j = tid; j < len; j += BLOCK) {
        unsigned voff = (unsigned)(e0 + j) * 4u;
        unsigned lds  = (unsigned)(unsigned long long)(&ds_raw[j]);
        asm volatile("global_load_async_to_lds_b32 %0, %1, %2"
                     :: "v"(lds), "v"(voff), "s"(csr_src) : "memory");
    }
    asm volatile("s_wait_asynccnt 0x0" ::: "memory");
    __syncthreads();

    // build packed records: src|slot<<22 , scale ; pad 8 entries (slot=32, scale=0)
    for (int j = tid; j < len + 8; j += BLOCK) {
        int meta = 32 << 22, scb = 0;
        if (j < len) {
            int s = ds_raw[j];
            int e = e0 + j, lo = 0, hi = NODES_PER_BLOCK;
            while (hi - lo > 1) {
                int mid = (lo + hi) >> 1;
                if (e >= ds_rs[mid]) lo = mid; else hi = mid;
            }
            meta = s | (lo << 22);
            scb  = __float_as_int(dis[s]);
        }
        *(int2*)(ds_rec + 2 * j) = make_int2(meta, scb);
    }
    __syncthreads();

    const int wave = tid >> 5;
    const int lane = tid & 31;
    const int g    = wave >> 2;            // node group (0/1)
    const int q    = wave & 3;             // feature quarter
    const int fb   = q << 4;
    const int half = lane >> 4;            // A: K-pair select; B: row-pair select
    const int mn   = lane & 15;            // m (feature) for A, n (node col) for B/D
    const unsigned laneByte = (unsigned)(fb + mn) << 2;
    const int myslot = 16 * g + mn;

    int gbeg = ds_rs[16 * g]      - e0; if (gbeg > len) gbeg = len;
    int gend = ds_rs[16 * g + 16] - e0; if (gend > len) gend = len;

    const char* fpb = (const char*)fprev;

    // software pipeline: rr = current chunk's 2 records, rn = next chunk's
    int k0 = gbeg + 2 * half;
    int4 rr = *(const int4*)(ds_rec + 2 * k0);     // pad makes this always safe
    v8f c = {0.f, 0.f, 0.f, 0.f, 0.f, 0.f, 0.f, 0.f};
    for (int j = gbeg; j < gend; j += 4) {
        int4 rn = *(const int4*)(ds_rec + 2 * (k0 + 4));
        // near-prefetch next chunk's gather rows
        __builtin_prefetch(fpb + (((unsigned)(rn.x & SRCM) << 8) + laneByte), 0, 3);
        __builtin_prefetch(fpb + (((unsigned)(rn.z & SRCM) << 8) + laneByte), 0, 3);

        v2f a, b;
        a.x = *(const float*)(fpb + (((unsigned)(rr.x & SRCM) << 8) + laneByte));
        a.y = *(const float*)(fpb + (((unsigned)(rr.z & SRCM) << 8) + laneByte));
        b.x = ((rr.x >> 22) == myslot) ? __int_as_float(rr.y) : 0.f;
        b.y = ((rr.z >> 22) == myslot) ? __int_as_float(rr.w) : 0.f;
        c = __builtin_amdgcn_wmma_f32_16x16x4_f32(false, a, false, b,
                                                  (short)0, c, false, false);
        rr = rn;
        k0 += 4;
    }

    // D layout: lanes 0-15 -> features fb+0..7, lanes 16-31 -> fb+8..15,
    // node column n0+16g+(lane&15); 8 contiguous floats per lane.
    const int node = n0 + 16 * g + mn;
    if (node < N_) {
        const int fo = fb + (half ? 8 : 0);
        const float di = dis[node];
        const float4* pp = (const float4*)(fprev + (size_t)node * FDIM + fo);
        float4* fc = (float4*)(fcur + (size_t)node * FDIM + fo);
        float4* op = (float4*)(out  + (size_t)node * FDIM + fo);
        float4 p0 = pp[0], p1 = pp[1];
        float4 o0 = op[0], o1 = op[1];
        float4 fn0, fn1;
        fn0.x = fmaf(-di, c[0], p0.x);  fn0.y = fmaf(-di, c[1], p0.y);
        fn0.z = fmaf(-di, c[2], p0.z);  fn0.w = fmaf(-di, c[3], p0.w);
        fn1.x = fmaf(-di, c[4], p1.x);  fn1.y = fmaf(-di, c[5], p1.y);
        fn1.z = fmaf(-di, c[6], p1.z);  fn1.w = fmaf(-di, c[7], p1.w);
        fc[0] = fn0;  fc[1] = fn1;
        o0.x = fmaf(theta, fn0.x, o0.x);  o0.y = fmaf(theta, fn0.y, o0.y);
        o0.z = fmaf(theta, fn0.z, o0.z);  o0.w = fmaf(theta, fn0.w, o0.w);
        o1.x = fmaf(theta, fn1.x, o1.x);  o1.y = fmaf(theta, fn1.y, o1.y);
        o1.z = fmaf(theta, fn1.z, o1.z);  o1.w = fmaf(theta, fn1.w, o1.w);
        op[0] = o0;  op[1] = o1;
    }
}

// ---------------------------------------------------------------------------
extern "C" void kernel_launch(void* const* d_in, const int* in_sizes, int n_in,
                              void* d_out, int out_size, void* d_ws, size_t ws_size,
                              hipStream_t stream) {
    const float* x    = (const float*)d_in[0];   // [N, 64]
    const int*   eidx = (const int*)d_in[1];     // [2, E]
    float*       out  = (float*)d_out;           // [N, 64]

    const int NF  = in_sizes[0];
    const int E   = in_sizes[1] / 2;
    const int N   = NF / FDIM;
    const int NF4 = NF / 4;

    int*   deg      = (int*)d_ws;           // [N]
    int*   rowstart = deg + N;              // [N+4]
    int*   cursor   = rowstart + N + 4;     // [N]
    float* dis      = (float*)(cursor + N); // [N]
    int*   csr_src  = (int*)(dis + N);      // [E]
    float* bufA     = (float*)(csr_src + E);
    float* bufB     = bufA + NF;

    static const float THETA[5] = {0.6f, -0.4f, 0.3f, -0.2f, 0.1f};

    const int gN   = (N + BLOCK - 1) / BLOCK;
    const int gE   = (E + BLOCK - 1) / BLOCK;
    const int gNF4 = (NF4 + BLOCK - 1) / BLOCK;
    const int gP   = (N + NODES_PER_BLOCK - 1) / NODES_PER_BLOCK;

    poly_zero_int<<<gN, BLOCK, 0, stream>>>(deg, N);
    poly_degree  <<<gE, BLOCK, 0, stream>>>(eidx, deg, E);
    poly_scan    <<<1, 1024, 0, stream>>>(deg, rowstart, N);
    poly_prep    <<<gN, BLOCK, 0, stream>>>(deg, rowstart, dis, cursor, N);
    poly_build   <<<gE, BLOCK, 0, stream>>>(eidx, cursor, csr_src, E);

    poly_init_out<<<gNF4, BLOCK, 0, stream>>>((const float4*)x, (float4*)out,
                                              THETA[0], NF4);

    const float* fprev = x;
    float* fcur = bufA;
    float* other = bufB;
    for (int k = 1; k <= 4; ++k) {
        poly_pass<<<gP, BLOCK, 0, stream>>>(rowstart, csr_src, dis,
                                            fprev, fcur, out, THETA[k], N);
        fprev = fcur;
        float* t = fcur; fcur = other; other = t;
    }
}